// LSTMScratch_10548439679037
// MI455X (gfx1250) — compile-verified
//
#include <hip/hip_runtime.h>

typedef float v2f __attribute__((ext_vector_type(2)));
typedef float v8f __attribute__((ext_vector_type(8)));

#define T_LEN 1024
#define B_SZ  256
#define D_SZ  256
#define H_SZ  256

// Zero the recurrent state (c) and the h0 buffer every call (determinism:
// harness poisons ws once; previous calls leave c_T behind).
__global__ void lstm_init_state(float* __restrict__ c, float* __restrict__ h0) {
    int idx = blockIdx.x * blockDim.x + threadIdx.x;
    if (idx < B_SZ * H_SZ) { c[idx] = 0.0f; h0[idx] = 0.0f; }
}

__device__ __forceinline__ float sigmoidf_dev(float x) {
    return 1.0f / (1.0f + __expf(-x));
}

// One LSTM timestep. Grid (H/16, B/16), block = 128 threads = 4 wave32s.
// Wave g computes gate g's 16x16 tile: acc = x_t(16xK) @ Wx(KxN) + h(16xK) @ Wh(KxN) + b
// using chained V_WMMA_F32_16X16X4_F32 (full fp32, K=512 total -> 128 WMMA/wave).
__global__ __launch_bounds__(128)
void lstm_step(const float* __restrict__ X,
               const float* __restrict__ Wxi, const float* __restrict__ Whi, const float* __restrict__ bi,
               const float* __restrict__ Wxf, const float* __restrict__ Whf, const float* __restrict__ bf,
               const float* __restrict__ Wxo, const float* __restrict__ Who, const float* __restrict__ bo,
               const float* __restrict__ Wxc, const float* __restrict__ Whc, const float* __restrict__ bc,
               const float* __restrict__ Hprev,   // [B,H] h_{t-1}
               float* __restrict__ Cstate,        // [B,H] c state (read/modify/write)
               float* __restrict__ Hout,          // [B,H] slice of outputs for this t
               float* __restrict__ HT,            // [B,H] final h (written only at t==T-1)
               float* __restrict__ CT,            // [B,H] final c (written only at t==T-1)
               int t)
{
    __shared__ float lds[4][16][17];   // +1 pad to dodge bank conflicts

    const int wave = threadIdx.x >> 5;   // gate id: 0=i 1=f 2=o 3=c~
    const int lane = threadIdx.x & 31;
    const int m    = lane & 15;          // A-row / B-col / D-col lane index
    const int kh   = lane >> 4;          // K-half select (lanes 16-31 hold K=2,3)
    const int col0 = blockIdx.x << 4;    // H tile origin
    const int row0 = blockIdx.y << 4;    // B tile origin

    const float* Wx; const float* Wh; const float* bv;
    switch (wave) {
        case 0:  Wx = Wxi; Wh = Whi; bv = bi; break;
        case 1:  Wx = Wxf; Wh = Whf; bv = bf; break;
        case 2:  Wx = Wxo; Wh = Who; bv = bo; break;
        default: Wx = Wxc; Wh = Whc; bv = bc; break;
    }

    // A-matrix rows this lane feeds (ISA 16x4 f32 A layout).
    const float* xrow = X + ((size_t)t * B_SZ + (size_t)(row0 + m)) * D_SZ;
    const float* hrow = Hprev + (size_t)(row0 + m) * H_SZ;

    v8f acc = {};   // 16x16 f32 accumulator (8 VGPRs)

    // ---- x_t @ Wx : K = D = 256, step 4 -> 64 WMMA ----
#pragma unroll 4
    for (int k = 0; k < D_SZ; k += 4) {
        const int kk = k + 2 * kh;                 // this lane's K pair
        v2f a = *(const v2f*)(xrow + kk);          // A[m][kk], A[m][kk+1]
        v2f b;
        b.x = Wx[(size_t)kk       * H_SZ + col0 + m];   // B[kk  ][n=m]
        b.y = Wx[(size_t)(kk + 1) * H_SZ + col0 + m];   // B[kk+1][n=m]
        acc = __builtin_amdgcn_wmma_f32_16x16x4_f32(false, a, false, b,
                                                    (short)0, acc, false, false);
    }

    // ---- h_{t-1} @ Wh : K = H = 256, step 4 -> 64 WMMA ----
#pragma unroll 4
    for (int k = 0; k < H_SZ; k += 4) {
        const int kk = k + 2 * kh;
        v2f a = *(const v2f*)(hrow + kk);
        v2f b;
        b.x = Wh[(size_t)kk       * H_SZ + col0 + m];
        b.y = Wh[(size_t)(kk + 1) * H_SZ + col0 + m];
        acc = __builtin_amdgcn_wmma_f32_16x16x4_f32(false, a, false, b,
                                                    (short)0, acc, false, false);
    }

    // Bias depends only on output column -> same for all 8 accumulator rows.
    const float bb = bv[col0 + m];

    // Spill this gate's tile to LDS (C/D layout: VGPR r holds M=r / M=r+8).
#pragma unroll
    for (int r = 0; r < 8; ++r) {
        lds[wave][r + 8 * kh][m] = acc[r] + bb;
    }
    __syncthreads();

    // Gating: 128 threads cover 256 tile elements.
    const bool last = (t == T_LEN - 1);
    for (int e = threadIdx.x; e < 256; e += 128) {
        const int mm = e >> 4;
        const int nn = e & 15;
        const float gi = sigmoidf_dev(lds[0][mm][nn]);
        const float gf = sigmoidf_dev(lds[1][mm][nn]);
        const float go = sigmoidf_dev(lds[2][mm][nn]);
        const float gc = tanhf(lds[3][mm][nn]);
        const size_t idx = (size_t)(row0 + mm) * H_SZ + (size_t)(col0 + nn);
        const float cn = gf * Cstate[idx] + gi * gc;
        const float hn = go * tanhf(cn);
        Cstate[idx] = cn;
        Hout[idx]   = hn;
        if (last) { HT[idx] = hn; CT[idx] = cn; }
    }
}

extern "C" void kernel_launch(void* const* d_in, const int* in_sizes, int n_in,
                              void* d_out, int out_size, void* d_ws, size_t ws_size,
                              hipStream_t stream) {
    (void)in_sizes; (void)n_in; (void)out_size; (void)ws_size;

    const float* X   = (const float*)d_in[0];
    const float* Wxi = (const float*)d_in[1];
    const float* Whi = (const float*)d_in[2];
    const float* bi  = (const float*)d_in[3];
    const float* Wxf = (const float*)d_in[4];
    const float* Whf = (const float*)d_in[5];
    const float* bf  = (const float*)d_in[6];
    const float* Wxo = (const float*)d_in[7];
    const float* Who = (const float*)d_in[8];
    const float* bo  = (const float*)d_in[9];
    const float* Wxc = (const float*)d_in[10];
    const float* Whc = (const float*)d_in[11];
    const float* bc  = (const float*)d_in[12];

    float* out  = (float*)d_out;
    float* Hseq = out;                                        // [T,B,H]
    float* HT   = out + (size_t)T_LEN * B_SZ * H_SZ;          // [B,H]
    float* CT   = HT + (size_t)B_SZ * H_SZ;                   // [B,H]

    float* C  = (float*)d_ws;                                 // [B,H] c state
    float* H0 = C + (size_t)B_SZ * H_SZ;                      // [B,H] zeros (h_{-1})

    lstm_init_state<<<(B_SZ * H_SZ + 255) / 256, 256, 0, stream>>>(C, H0);

    dim3 grid(H_SZ / 16, B_SZ / 16);
    for (int t = 0; t < T_LEN; ++t) {
        const float* Hprev = (t == 0) ? H0 : (Hseq + (size_t)(t - 1) * B_SZ * H_SZ);
        lstm_step<<<grid, 128, 0, stream>>>(X,
                                            Wxi, Whi, bi,
                                            Wxf, Whf, bf,
                                            Wxo, Who, bo,
                                            Wxc, Whc, bc,
                                            Hprev, C,
                                            Hseq + (size_t)t * B_SZ * H_SZ,
                                            HT, CT, t);
    }
}